// CostVolume_1726576855665
// MI455X (gfx1250) — compile-verified
//
#include <hip/hip_runtime.h>
#include <stdint.h>

typedef __attribute__((ext_vector_type(2))) float v2f;
typedef __attribute__((ext_vector_type(8))) float v8f;

#define C_DIM 128
#define H_DIM 160
#define W_DIM 512
#define D_DIM 64
#define CHUNK 32          // channels staged in LDS per pipeline stage
#define JBLK  64          // j-columns per workgroup (4 waves x 16)
#define CPAD  36          // LDS row stride (floats): 144B rows -> conflict-free b64 frags

// HBM -> LDS direct async copy, 4 bytes per lane, per-lane LDS destination.
// GVS addressing (saddr64 + vaddr32); tracked by ASYNCcnt; EXEC-masked lanes skip.
__device__ __forceinline__ void async_cp_b32(unsigned lds_addr, int gbyte_off,
                                             unsigned long long gbase) {
    asm volatile("global_load_async_to_lds_b32 %0, %1, %2"
                 :: "v"(lds_addr), "v"(gbyte_off), "s"(gbase)
                 : "memory");
}

__device__ __forceinline__ void wait_async0() {
    asm volatile("s_wait_asynccnt 0x0" ::: "memory");
}

// One workgroup = one (b,h) slice x 64 output columns.
// Wave wv owns j-tile j0w = wv*16 and 5 diagonal 16x16 k-tiles covering d in [0,64):
// cost[d,j] = G[j, j-d]/C with G = L^T R, computed as banded WMMA f32 16x16x4 GEMM.
// LDS is stored TRANSPOSED ([column][channel]) so every WMMA operand is one
// aligned ds_load_b64 -> even VGPR pair, fed straight into v_wmma.
__global__ __launch_bounds__(128) void cost_volume_wmma_f32(
    const float* __restrict__ left,
    const float* __restrict__ right,
    float* __restrict__ out)
{
    __shared__ float Lt[2][JBLK][CPAD];        // Lt[buf][j - base][c]
    __shared__ float Rt[2][2 * JBLK][CPAD];    // Rt[buf][k - (base-64)][c]; k<0 rows stay 0

    const int t    = threadIdx.x;
    const int lane = t & 31;
    const int wv   = t >> 5;

    int bx = blockIdx.x;
    const int jb = bx & 7;                     // W / JBLK = 8
    bx >>= 3;
    const int hh = bx % H_DIM;
    const int bb = bx / H_DIM;
    const int base = jb * JBLK;

    const size_t cstride = (size_t)H_DIM * W_DIM;   // element stride between channels
    const float* Lp = left  + (size_t)bb * C_DIM * cstride + (size_t)hh * W_DIM;
    const float* Rp = right + (size_t)bb * C_DIM * cstride + (size_t)hh * W_DIM;
    const unsigned long long LpA = (unsigned long long)(uintptr_t)Lp;
    const unsigned long long RpA = (unsigned long long)(uintptr_t)Rp;

    const int j0w    = wv * 16;
    const int halfHi = (lane >> 4);            // 0: lanes 0-15, 1: lanes 16-31
    const int nIdx   = lane & 15;

    // ---- pre-zero the k<0 rows of both R buffers (left image edge only) ----
    if (base == 0) {
        const float4 z4 = make_float4(0.f, 0.f, 0.f, 0.f);
        #pragma unroll
        for (int i = 0; i < 9; ++i) {          // 2 bufs * 64 rows * 9 float4 = 1152
            const int f    = t + i * 128;
            const int bufz = f / 576;
            const int fr   = f % 576;
            const int row  = fr / 9;
            const int col4 = (fr % 9) * 4;
            *(float4*)&Rt[bufz][row][col4] = z4;
        }
    }

    // ---- async stage: HBM -> transposed LDS, 48 b32 instrs / wave / stage ----
    auto stage = [&](int buf, int cc) {
        #pragma unroll
        for (int i = 0; i < 16; ++i) {         // L: 2048 elems, lanes sweep j (coalesced)
            const int gidx = ((wv << 4) + i) * 32 + lane;
            const int c = gidx >> 6;           // uniform within one instruction
            const int j = gidx & 63;
            const unsigned lds = (unsigned)(uintptr_t)&Lt[buf][j][c];
            const int goff = ((cc + c) * (int)cstride + base + j) * 4;
            async_cp_b32(lds, goff, LpA);
        }
        #pragma unroll
        for (int i = 0; i < 32; ++i) {         // R: 4096 elems
            const int gidx = ((wv << 5) + i) * 32 + lane;
            const int c  = gidx >> 7;          // uniform within one instruction
            const int jk = gidx & 127;
            const int gc = base - 64 + jk;
            if (gc >= 0) {
                const unsigned lds = (unsigned)(uintptr_t)&Rt[buf][jk][c];
                const int goff = ((cc + c) * (int)cstride + gc) * 4;
                async_cp_b32(lds, goff, RpA);
            }
        }
    };

    v8f acc[5];
    #pragma unroll
    for (int b5 = 0; b5 < 5; ++b5) acc[b5] = (v8f){0.f,0.f,0.f,0.f,0.f,0.f,0.f,0.f};

    stage(0, 0);                               // prime the pipeline

    for (int cc = 0; cc < C_DIM; cc += CHUNK) {
        const int buf = (cc >> 5) & 1;
        wait_async0();                         // this buffer's transfers landed
        __syncthreads();                       // ...for every wave in the group
        if (cc + CHUNK < C_DIM)
            stage(buf ^ 1, cc + CHUNK);        // overlap next transfer with compute

        // ---- 8 K-steps of 4, 5 band tiles each: 40 WMMA per chunk ----
        #pragma unroll
        for (int kk = 0; kk < CHUNK; kk += 4) {
            // A[M=j-j0, K=c-kk]: VGPR0 = K 0/2 per lane half, VGPR1 = K 1/3.
            // Transposed LDS makes each operand one aligned b64 load.
            const int cA = kk + (halfHi << 1);
            const v2f a = *(const v2f*)&Lt[buf][j0w + nIdx][cA];
            #pragma unroll
            for (int bnd = 0; bnd < 5; ++bnd) {
                const v2f bf = *(const v2f*)&Rt[buf][j0w + 16 * bnd + nIdx][cA];
                acc[bnd] = __builtin_amdgcn_wmma_f32_16x16x4_f32(
                    false, a, false, bf, (short)0, acc[bnd], false, false);
            }
        }
        __syncthreads();                       // done reading before refill next round
    }

    // ---- scatter D tiles into cost[b, d, h, j]; each (d,j) written exactly once ----
    const float scale = 1.0f / (float)C_DIM;
    float* Op = out + ((size_t)bb * D_DIM * H_DIM + hh) * (size_t)W_DIM;
    const int j0g = base + j0w;
    #pragma unroll
    for (int bnd = 0; bnd < 5; ++bnd) {
        const int kg = (j0g - 64 + 16 * bnd) + nIdx;
        #pragma unroll
        for (int r = 0; r < 8; ++r) {
            const int jg = j0g + r + (halfHi << 3);   // D layout: M=r / M=r+8
            const int d  = jg - kg;
            if (d >= 0 && d < D_DIM) {
                Op[(size_t)d * cstride + jg] = acc[bnd][r] * scale;
            }
        }
    }
}

extern "C" void kernel_launch(void* const* d_in, const int* in_sizes, int n_in,
                              void* d_out, int out_size, void* d_ws, size_t ws_size,
                              hipStream_t stream) {
    (void)in_sizes; (void)n_in; (void)d_ws; (void)ws_size; (void)out_size;
    const float* left  = (const float*)d_in[0];
    const float* right = (const float*)d_in[1];
    float* out = (float*)d_out;

    const int B = 8;
    const int grid = B * H_DIM * (W_DIM / JBLK);   // 10240 workgroups
    cost_volume_wmma_f32<<<grid, 128, 0, stream>>>(left, right, out);
}